// RelationMessagePassingBase_64536178589837
// MI455X (gfx1250) — compile-verified
//
#include <hip/hip_runtime.h>
#include <cstddef>
#include <cstdint>

typedef float v2f __attribute__((ext_vector_type(2)));
typedef float v8f __attribute__((ext_vector_type(8)));

__device__ __forceinline__ float mish_f(float x) {
    // x * tanh(softplus(x)), stable for large |x|
    float sp = (x > 20.0f) ? x : log1pf(__expf(x));
    return x * tanhf(sp);
}

// ---- gfx1250 async global->LDS copies (ASYNCcnt-tracked) -------------------
__device__ __forceinline__ void async_g2l_b128(uint32_t lds_addr, const void* gptr) {
    asm volatile("global_load_async_to_lds_b128 %0, %1, off"
                 :: "v"(lds_addr), "v"((unsigned long long)(uintptr_t)gptr)
                 : "memory");
}
__device__ __forceinline__ void async_g2l_b32(uint32_t lds_addr, const void* gptr) {
    asm volatile("global_load_async_to_lds_b32 %0, %1, off"
                 :: "v"(lds_addr), "v"((unsigned long long)(uintptr_t)gptr)
                 : "memory");
}
__device__ __forceinline__ void wait_async0() {
    asm volatile("s_wait_asynccnt 0x0" ::: "memory");
}

// One relation: out[t*D + c] = X[t,c] + (mish(X@W1 + b1) @ W2 + b2)[t,c]
// X gathered: X[t, p*64 + e] = emb[idx[t*A + p]][e]
//
// LDS W layout is K-pair interleaved: element (k,c) at (k>>1)*PS + 2c + (k&1),
// so a B fragment {W[ka][c], W[ka+1][c]} is one contiguous ds_load_b64.
template<int A>
__global__ __launch_bounds__(128)
void rel_mlp_kernel(const float* __restrict__ emb,
                    const int*   __restrict__ idx,
                    const float* __restrict__ W1, const float* __restrict__ pb1,
                    const float* __restrict__ W2, const float* __restrict__ pb2,
                    float* __restrict__ out, int n_tuples)
{
    constexpr int D  = 64 * A;
    constexpr int PS = 2 * D + 32;  // interleaved-W pair-row stride (PS%64==32)
    constexpr int XS = D + 4;       // X/H row stride (conflict-free A frags)

    extern __shared__ float smem[];
    float* sW = smem;                    // (D/2) * PS
    float* sX = smem + (D / 2) * PS;     // 64 * XS
    float* sH = sX + 64 * XS;            // 64 * XS

    const uint32_t sW_a = (uint32_t)(uintptr_t)sW;
    const uint32_t sX_a = (uint32_t)(uintptr_t)sX;

    const int tid  = threadIdx.x;       // 0..127
    const int wave = tid >> 5;          // 0..3
    const int lane = tid & 31;
    const int m    = lane & 15;         // M (A-frag) / N (B,C-frag)
    const int half = lane >> 4;         // K-half for A/B frags, row-half for C
    const int row0 = blockIdx.x * 64;   // first tuple of this tile
    const int mrow = wave * 16 + m;     // A-fragment row within tile

    // ---- async gather of X tile: 64 rows x D floats, b128 per lane ----
    {
        constexpr int NF4 = D / 4;      // float4 per row
        for (int i = tid; i < 64 * NF4; i += 128) {   // multiple of 128: EXEC full
            const int r  = i / NF4;
            const int c4 = i - r * NF4;
            int t = row0 + r; if (t >= n_tuples) t = n_tuples - 1;
            const int g = idx[t * A + (c4 >> 4)];
            async_g2l_b128(sX_a + (uint32_t)(r * XS + c4 * 4) * 4u,
                           emb + (size_t)g * 64 + (c4 & 15) * 4);
        }
    }
    // ---- async stage W1 into interleaved LDS layout (b32 per element) ----
    for (int pr = 0; pr < D / 2; ++pr) {
        const float*   grow  = W1 + (size_t)(2 * pr) * D;     // rows 2pr, 2pr+1
        const uint32_t lbase = sW_a + (uint32_t)(pr * PS) * 4u;
#pragma unroll
        for (int u = 0; u < A; ++u) {
            const int o  = u * 128 + tid;    // 0 .. 2D-1
            const int c  = o >> 1;
            const int rb = o & 1;
            async_g2l_b32(lbase + (uint32_t)o * 4u, grow + (size_t)rb * D + c);
        }
    }
    wait_async0();
    __syncthreads();

    // ---- GEMM1: H = mish(X @ W1 + b1), 4 column tiles per A fragment ----
    for (int nt = 0; nt < D / 16; nt += 4) {
        const int c0 = nt * 16 + m;
        v8f acc0 = {}, acc1 = {}, acc2 = {}, acc3 = {};
        for (int k = 0; k < D; k += 4) {
            const int ka  = k + 2 * half;
            const int pro = (ka >> 1) * PS;
            const v2f a   = *reinterpret_cast<const v2f*>(&sX[mrow * XS + ka]);
            const v2f bf0 = *reinterpret_cast<const v2f*>(&sW[pro + 2 * c0]);
            const v2f bf1 = *reinterpret_cast<const v2f*>(&sW[pro + 2 * (c0 + 16)]);
            const v2f bf2 = *reinterpret_cast<const v2f*>(&sW[pro + 2 * (c0 + 32)]);
            const v2f bf3 = *reinterpret_cast<const v2f*>(&sW[pro + 2 * (c0 + 48)]);
            acc0 = __builtin_amdgcn_wmma_f32_16x16x4_f32(false, a, false, bf0, (short)0, acc0, false, false);
            acc1 = __builtin_amdgcn_wmma_f32_16x16x4_f32(false, a, false, bf1, (short)0, acc1, false, false);
            acc2 = __builtin_amdgcn_wmma_f32_16x16x4_f32(false, a, false, bf2, (short)0, acc2, false, false);
            acc3 = __builtin_amdgcn_wmma_f32_16x16x4_f32(false, a, false, bf3, (short)0, acc3, false, false);
        }
        const float bb0 = pb1[c0];
        const float bb1 = pb1[c0 + 16];
        const float bb2 = pb1[c0 + 32];
        const float bb3 = pb1[c0 + 48];
#pragma unroll
        for (int r = 0; r < 8; ++r) {
            const int rr = r + 8 * half;                 // C-matrix row within 16
            float* hrow = &sH[(wave * 16 + rr) * XS];
            hrow[c0]      = mish_f(acc0[r] + bb0);
            hrow[c0 + 16] = mish_f(acc1[r] + bb1);
            hrow[c0 + 32] = mish_f(acc2[r] + bb2);
            hrow[c0 + 48] = mish_f(acc3[r] + bb3);
        }
    }
    __syncthreads();   // all GEMM1 reads of sW done before overwrite

    // ---- async stage W2 into interleaved LDS (reuse sW) ----
    for (int pr = 0; pr < D / 2; ++pr) {
        const float*   grow  = W2 + (size_t)(2 * pr) * D;
        const uint32_t lbase = sW_a + (uint32_t)(pr * PS) * 4u;
#pragma unroll
        for (int u = 0; u < A; ++u) {
            const int o  = u * 128 + tid;
            const int c  = o >> 1;
            const int rb = o & 1;
            async_g2l_b32(lbase + (uint32_t)o * 4u, grow + (size_t)rb * D + c);
        }
    }
    wait_async0();
    __syncthreads();

    // ---- GEMM2: Y = X + H @ W2 + b2, store contiguous ----
    for (int nt = 0; nt < D / 16; nt += 4) {
        const int c0 = nt * 16 + m;
        v8f acc0 = {}, acc1 = {}, acc2 = {}, acc3 = {};
        for (int k = 0; k < D; k += 4) {
            const int ka  = k + 2 * half;
            const int pro = (ka >> 1) * PS;
            const v2f a   = *reinterpret_cast<const v2f*>(&sH[mrow * XS + ka]);
            const v2f bf0 = *reinterpret_cast<const v2f*>(&sW[pro + 2 * c0]);
            const v2f bf1 = *reinterpret_cast<const v2f*>(&sW[pro + 2 * (c0 + 16)]);
            const v2f bf2 = *reinterpret_cast<const v2f*>(&sW[pro + 2 * (c0 + 32)]);
            const v2f bf3 = *reinterpret_cast<const v2f*>(&sW[pro + 2 * (c0 + 48)]);
            acc0 = __builtin_amdgcn_wmma_f32_16x16x4_f32(false, a, false, bf0, (short)0, acc0, false, false);
            acc1 = __builtin_amdgcn_wmma_f32_16x16x4_f32(false, a, false, bf1, (short)0, acc1, false, false);
            acc2 = __builtin_amdgcn_wmma_f32_16x16x4_f32(false, a, false, bf2, (short)0, acc2, false, false);
            acc3 = __builtin_amdgcn_wmma_f32_16x16x4_f32(false, a, false, bf3, (short)0, acc3, false, false);
        }
        const float bb0 = pb2[c0];
        const float bb1 = pb2[c0 + 16];
        const float bb2 = pb2[c0 + 32];
        const float bb3 = pb2[c0 + 48];
#pragma unroll
        for (int r = 0; r < 8; ++r) {
            const int rr   = r + 8 * half;
            const int trow = row0 + wave * 16 + rr;
            if (trow < n_tuples) {
                const float* xrow = &sX[(wave * 16 + rr) * XS];
                float* orow = out + (size_t)trow * D;
                orow[c0]      = acc0[r] + bb0 + xrow[c0];
                orow[c0 + 16] = acc1[r] + bb1 + xrow[c0 + 16];
                orow[c0 + 32] = acc2[r] + bb2 + xrow[c0 + 32];
                orow[c0 + 48] = acc3[r] + bb3 + xrow[c0 + 48];
            }
        }
    }
}

static inline size_t smem_bytes(int a) {
    const int D = 64 * a;
    const int PS = 2 * D + 32;
    const int XS = D + 4;
    return ((size_t)(D / 2) * PS + 2u * 64u * XS) * sizeof(float);
}

extern "C" void kernel_launch(void* const* d_in, const int* in_sizes, int n_in,
                              void* d_out, int out_size, void* d_ws, size_t ws_size,
                              hipStream_t stream) {
    (void)n_in; (void)out_size; (void)d_ws; (void)ws_size;
    static const int ARITY[8] = {1, 2, 2, 2, 3, 3, 2, 1};

    const float* emb = (const float*)d_in[0];
    const int NT = in_sizes[1];          // rel0 has arity 1 -> NT tuples

    // Raise dynamic-LDS caps (max 254 KB for D=192; CDNA5 WGP allows 320 KB)
    hipFuncSetAttribute(reinterpret_cast<const void*>(rel_mlp_kernel<1>),
                        hipFuncAttributeMaxDynamicSharedMemorySize, (int)smem_bytes(1));
    hipFuncSetAttribute(reinterpret_cast<const void*>(rel_mlp_kernel<2>),
                        hipFuncAttributeMaxDynamicSharedMemorySize, (int)smem_bytes(2));
    hipFuncSetAttribute(reinterpret_cast<const void*>(rel_mlp_kernel<3>),
                        hipFuncAttributeMaxDynamicSharedMemorySize, (int)smem_bytes(3));

    float* out_msgs = (float*)d_out;
    size_t total_rows = 0;
    for (int r = 0; r < 8; ++r) total_rows += (size_t)NT * ARITY[r];
    int* out_idx = reinterpret_cast<int*>(out_msgs + total_rows * 64);

    size_t msg_row = 0;
    for (int r = 0; r < 8; ++r) {
        const int a = ARITY[r];
        const int* ix = (const int*)d_in[1 + r];
        // params dict insertion order per relation: W1, b1, W2, b2
        const float* w1 = (const float*)d_in[9 + 4 * r + 0];
        const float* b1 = (const float*)d_in[9 + 4 * r + 1];
        const float* w2 = (const float*)d_in[9 + 4 * r + 2];
        const float* b2 = (const float*)d_in[9 + 4 * r + 3];
        float* o = out_msgs + msg_row * 64;

        const int nb = (NT + 63) / 64;
        const size_t smem = smem_bytes(a);
        if (a == 1) {
            rel_mlp_kernel<1><<<nb, 128, smem, stream>>>(emb, ix, w1, b1, w2, b2, o, NT);
        } else if (a == 2) {
            rel_mlp_kernel<2><<<nb, 128, smem, stream>>>(emb, ix, w1, b1, w2, b2, o, NT);
        } else {
            rel_mlp_kernel<3><<<nb, 128, smem, stream>>>(emb, ix, w1, b1, w2, b2, o, NT);
        }
        hipMemcpyAsync(out_idx + msg_row, ix, (size_t)NT * a * sizeof(int),
                       hipMemcpyDeviceToDevice, stream);
        msg_row += (size_t)NT * a;
    }
}